// PNA_11424613007592
// MI455X (gfx1250) — compile-verified
//
#include <hip/hip_runtime.h>
#include <hip/hip_bf16.h>
#include <math.h>

// Problem constants (from the reference)
#define N_NODES 50000
#define E_EDGES 800000
#define IN_DIM  128
#define H_DIM   64
#define T_TOW   2
#define FO_DIM  32          // H/T
#define K13     832         // 13*H per tower

typedef __attribute__((ext_vector_type(2))) float v2f;
typedef __attribute__((ext_vector_type(8))) float v8f;

// ---------------------------------------------------------------------------
// WMMA helpers: V_WMMA_F32_16X16X4_F32, wave32, one wave -> one 16x16 f32 tile
// A fragment (16x4, M x K): lane<16 holds (M=lane, K=k..k+1), lane>=16 holds
// (M=lane-16, K=k+2..k+3).  B (4x16, K x N) mirrors with N in place of M.
// D (16x16): VGPR r holds (M=r, N=lane) for lanes 0-15, (M=r+8) for 16-31.
// ---------------------------------------------------------------------------
__device__ __forceinline__ v8f wmma_step(float a0, float a1, float b0, float b1, v8f acc) {
  v2f a; a.x = a0; a.y = a1;
  v2f b; b.x = b0; b.y = b1;
  return __builtin_amdgcn_wmma_f32_16x16x4_f32(false, a, false, b, (short)0, acc,
                                               false, false);
}

// D(16x16) = A(16xK, row-major lda) * B(KxN>=16, row-major ldb, pre-offset to col 0)
__device__ __forceinline__ v8f wmma_gemm_tile(const float* A, int lda,
                                              const float* B, int ldb,
                                              int K, int lane) {
  const int half = lane >> 4;
  const int idx  = lane & 15;
  v8f acc = {};
  for (int k = 0; k < K; k += 4) {
    const int ka = k + half * 2;
    acc = wmma_step(A[(size_t)idx * lda + ka], A[(size_t)idx * lda + ka + 1],
                    B[(size_t)ka * ldb + idx], B[(size_t)(ka + 1) * ldb + idx], acc);
  }
  return acc;
}

// Store 16x16 tile (+ per-column bias); D pre-offset to column 0 of the tile.
__device__ __forceinline__ void wmma_store_tile(float* D, int ldd, v8f acc,
                                                const float* bias, int lane) {
  const int half = lane >> 4;
  const int idx  = lane & 15;
  const float bv = bias ? bias[idx] : 0.0f;
#pragma unroll
  for (int r = 0; r < 8; ++r)
    D[(size_t)(r + 8 * half) * ldd + idx] = acc[r] + bv;
}

// ---------------------------------------------------------------------------
// Kernels
// ---------------------------------------------------------------------------
__global__ void k_zero(float* p, size_t n) {
  size_t i = (size_t)blockIdx.x * blockDim.x + threadIdx.x;
  const size_t step = (size_t)gridDim.x * blockDim.x;
  for (; i < n; i += step) p[i] = 0.0f;
}

__global__ void k_degcount(const int* __restrict__ dst, int* __restrict__ degi) {
  int e = blockIdx.x * blockDim.x + threadIdx.x;
  if (e < E_EDGES) atomicAdd(&degi[dst[e]], 1);
}

// Single-block exclusive scan over N node degrees -> row_ptr (and cursor copy).
// N=50k, trivial cost; Hillis-Steele in LDS, 1024 threads, 49 chunks.
__global__ void __launch_bounds__(1024) k_scan(const int* __restrict__ degi,
                                               int* __restrict__ row_ptr,
                                               int* __restrict__ cursor) {
  __shared__ int buf[1024];
  __shared__ int carry;
  const int tid = threadIdx.x;
  if (tid == 0) carry = 0;
  __syncthreads();
  for (int base = 0; base < N_NODES; base += 1024) {
    const int n = base + tid;
    const int v = (n < N_NODES) ? degi[n] : 0;
    buf[tid] = v;
    __syncthreads();
    for (int off = 1; off < 1024; off <<= 1) {
      const int t = (tid >= off) ? buf[tid - off] : 0;
      __syncthreads();
      buf[tid] += t;
      __syncthreads();
    }
    const int c = carry;
    if (n < N_NODES) {
      const int excl = c + buf[tid] - v;
      row_ptr[n] = excl;
      cursor[n] = excl;
    }
    __syncthreads();
    if (tid == 1023) carry = c + buf[1023];
    __syncthreads();
  }
  if (tid == 0) row_ptr[N_NODES] = carry;  // == E
}

// Bucket edges by dst: col[pos] = src(e)
__global__ void k_fill(const int* __restrict__ src, const int* __restrict__ dst,
                       int* __restrict__ cursor, int* __restrict__ col) {
  int e = blockIdx.x * blockDim.x + threadIdx.x;
  if (e < E_EDGES) {
    const int pos = atomicAdd(&cursor[dst[e]], 1);
    col[pos] = src[e];
  }
}

__global__ void k_avglog(const int* __restrict__ row_ptr, float* __restrict__ logsum) {
  __shared__ float red[256];
  float s = 0.0f;
  for (int n = blockIdx.x * blockDim.x + threadIdx.x; n < N_NODES;
       n += gridDim.x * blockDim.x)
    s += logf((float)(row_ptr[n + 1] - row_ptr[n]) + 1.0f);
  red[threadIdx.x] = s;
  __syncthreads();
  for (int off = 128; off > 0; off >>= 1) {
    if ((int)threadIdx.x < off) red[threadIdx.x] += red[threadIdx.x + off];
    __syncthreads();
  }
  if (threadIdx.x == 0) atomicAdd(logsum, red[0]);
}

// h = x @ embed_W + embed_b     (N x 128)(128 x 64) ; 1 block = 16 nodes, 4 waves
__global__ void __launch_bounds__(128) k_embed(const float* __restrict__ x,
                                               const float* __restrict__ W,
                                               const float* __restrict__ b,
                                               float* __restrict__ h) {
  const int nb = blockIdx.x * 16;
  const int wid = threadIdx.x >> 5, lane = threadIdx.x & 31;
  const int colbase = wid * 16;
  v8f acc = wmma_gemm_tile(x + (size_t)nb * IN_DIM, IN_DIM, W + colbase, H_DIM,
                           IN_DIM, lane);
  wmma_store_tile(h + (size_t)nb * H_DIM + colbase, H_DIM, acc, b + colbase, lane);
}

// Ad[n, t*64+g] = h[n] @ preW[t][0:64]  (+ preb folded)   (dst-part of message)
// Bs[n, t*64+g] = h[n] @ preW[t][64:128]                  (src-part of message)
// grid.y: 0..3 -> (t, half);  1 block = 16 nodes, 4 waves = 64 cols
__global__ void __launch_bounds__(128) k_pre(const float* __restrict__ h,
                                             const float* __restrict__ preW,
                                             const float* __restrict__ preb,
                                             float* __restrict__ Ad,
                                             float* __restrict__ Bs) {
  const int nb = blockIdx.x * 16;
  const int t = blockIdx.y >> 1, hw = blockIdx.y & 1;
  const int wid = threadIdx.x >> 5, lane = threadIdx.x & 31;
  const int colbase = wid * 16;
  const float* B = preW + ((size_t)t * 128 + hw * 64) * H_DIM + colbase;
  v8f acc = wmma_gemm_tile(h + (size_t)nb * H_DIM, H_DIM, B, H_DIM, H_DIM, lane);
  float* outp = (hw ? Bs : Ad) + (size_t)nb * 128 + t * 64 + colbase;
  const float* bias = hw ? nullptr : (preb + t * 64 + colbase);
  wmma_store_tile(outp, 128, acc, bias, lane);
}

// Fused per 16-node tile:
//   CSR gather-reduce of Bs[src] stats (sum/sumsq/min/max, per tower) -> registers
//   analytic recombination with Ad (dst part, linearity of the message)
//   build [h | agg | agg*r1 | agg*r2] (832) in LDS
//   post GEMM (832x32 per tower, WMMA f32) + postb
//   lin GEMM (64x64, WMMA f32) + linb
// 128 threads (4 waves). Lanes of a wave read consecutive channels of the same
// source row -> 128B coalesced per edge; col[] is wave-uniform (L1 broadcast).
__global__ void __launch_bounds__(128) k_post(
    const float* __restrict__ h, const int* __restrict__ row_ptr,
    const int* __restrict__ col, const float* __restrict__ logsum,
    const float* __restrict__ Ad, const float* __restrict__ Bs,
    const float* __restrict__ postW, const float* __restrict__ postb,
    const float* __restrict__ linW, const float* __restrict__ linb,
    float* __restrict__ out) {
  __shared__ float Abuf[16][K13 + 8];   // 16 x 840 f32 = 53.8 KB
  __shared__ float Pout[16][72];        // post result (16 x 64) padded
  const int tid = threadIdx.x;
  const int wid = tid >> 5, lane = tid & 31;
  const int half = lane >> 4, idx = lane & 15;
  const int nb = blockIdx.x * 16;
  const float avg = logsum[0] * (1.0f / (float)N_NODES);

  // h block (cols 0..63), shared by both towers
  for (int q = tid; q < 1024; q += 128) {
    const int i = q >> 6, c = q & 63;
    Abuf[i][c] = h[(size_t)(nb + i) * H_DIM + c];
  }

  for (int tau = 0; tau < T_TOW; ++tau) {
    __syncthreads();  // previous tower's WMMA consumers done before overwrite
    for (int q = tid; q < 1024; q += 128) {
      const int i = q >> 6, g = q & 63;
      const int n = nb + i;
      const int beg = row_ptr[n], end = row_ptr[n + 1];
      const float dr = (float)(end - beg);
      const float cnt = fmaxf(dr, 1.0f);
      const float ld = logf(cnt + 1.0f);
      const float r1 = ld / avg, r2 = avg / ld;
      const size_t ch = (size_t)n * 128 + tau * 64 + g;
      const float A = Ad[ch];              // dst-part of message (incl. preb)
      float sB = 0.0f, sB2 = 0.0f;
      float mnB = 3.402823466e38f, mxB = -3.402823466e38f;
      for (int p = beg; p < end; ++p) {
        const int s = col[p];
        const float v = Bs[(size_t)s * 128 + tau * 64 + g];
        sB += v;
        sB2 += v * v;
        mnB = fminf(mnB, v);
        mxB = fmaxf(mxB, v);
      }
      const float inv = 1.0f / cnt;
      const float mean = (dr * A + sB) * inv;
      const float msq = (dr * A * A + 2.0f * A * sB + sB2) * inv;
      float mnv = 0.0f, mxv = 0.0f;
      if (end > beg) {
        mnv = A + mnB;
        mxv = A + mxB;
      }
      const float stdv = sqrtf(fmaxf(msq - mean * mean, 0.0f) + 1e-5f);
      float* row = Abuf[i];
      row[64 + g] = mean;       row[128 + g] = mnv;
      row[192 + g] = mxv;       row[256 + g] = stdv;
      row[320 + g] = mean * r1; row[384 + g] = mnv * r1;
      row[448 + g] = mxv * r1;  row[512 + g] = stdv * r1;
      row[576 + g] = mean * r2; row[640 + g] = mnv * r2;
      row[704 + g] = mxv * r2;  row[768 + g] = stdv * r2;
    }
    __syncthreads();
    if (wid < 2) {  // 2 waves x 16 cols = 32 outputs of this tower
      const float* Bp = postW + (size_t)tau * K13 * FO_DIM + wid * 16;
      __builtin_prefetch(Bp, 0, 1);
      v8f acc = {};
      for (int k = 0; k < K13; k += 4) {
        const int ka = k + half * 2;
        acc = wmma_step(Abuf[idx][ka], Abuf[idx][ka + 1],
                        Bp[(size_t)ka * FO_DIM + idx],
                        Bp[(size_t)(ka + 1) * FO_DIM + idx], acc);
      }
      const int gc = tau * FO_DIM + wid * 16 + idx;  // col of reshape(N, 64)
      const float bias = postb[gc];
#pragma unroll
      for (int r = 0; r < 8; ++r) Pout[r + 8 * half][gc] = acc[r] + bias;
    }
  }
  __syncthreads();

  // lin: (16 x 64) @ linW(64 x 64) + linb ; 4 waves x 16 cols
  {
    const int colbase = wid * 16;
    v8f acc = {};
    for (int k = 0; k < H_DIM; k += 4) {
      const int ka = k + half * 2;
      acc = wmma_step(Pout[idx][ka], Pout[idx][ka + 1],
                      linW[(size_t)ka * H_DIM + colbase + idx],
                      linW[(size_t)(ka + 1) * H_DIM + colbase + idx], acc);
    }
    const float bias = linb[colbase + idx];
#pragma unroll
    for (int r = 0; r < 8; ++r)
      out[(size_t)(nb + r + 8 * half) * H_DIM + colbase + idx] = acc[r] + bias;
  }
}

// BatchNorm stats: per-feature sum / sumsq via per-block partials + atomics
__global__ void __launch_bounds__(256) k_bnstat(const float* __restrict__ hout,
                                                float* __restrict__ bnsum,
                                                float* __restrict__ bnsq) {
  const int f = threadIdx.x & 63, sub = threadIdx.x >> 6;  // 4 node lanes
  float s = 0.0f, s2 = 0.0f;
  for (int n = blockIdx.x * 4 + sub; n < N_NODES; n += gridDim.x * 4) {
    const float v = hout[(size_t)n * H_DIM + f];
    s += v;
    s2 += v * v;
  }
  __shared__ float rs[256], rq[256];
  rs[threadIdx.x] = s;
  rq[threadIdx.x] = s2;
  __syncthreads();
  if (sub == 0) {
    for (int k = 1; k < 4; ++k) {
      s += rs[k * 64 + f];
      s2 += rq[k * 64 + f];
    }
    atomicAdd(&bnsum[f], s);
    atomicAdd(&bnsq[f], s2);
  }
}

__global__ void k_bnapply(const float* __restrict__ hout,
                          const float* __restrict__ bnsum,
                          const float* __restrict__ bnsq,
                          const float* __restrict__ gamma,
                          const float* __restrict__ beta, float* __restrict__ h) {
  size_t i = (size_t)blockIdx.x * blockDim.x + threadIdx.x;
  if (i >= (size_t)N_NODES * H_DIM) return;
  const int f = (int)(i & 63);
  const float mu = bnsum[f] * (1.0f / (float)N_NODES);
  const float var = bnsq[f] * (1.0f / (float)N_NODES) - mu * mu;
  const float inv = rsqrtf(var + 1e-5f);
  const float v = gamma[f] * (hout[i] - mu) * inv + beta[f];
  h[i] = fmaxf(v, 0.0f);
}

// ---------------------------------------------------------------------------
extern "C" void kernel_launch(void* const* d_in, const int* in_sizes, int n_in,
                              void* d_out, int out_size, void* d_ws, size_t ws_size,
                              hipStream_t stream) {
  (void)in_sizes; (void)n_in; (void)out_size; (void)ws_size;
  const float* x        = (const float*)d_in[0];
  const float* embed_W  = (const float*)d_in[1];
  const float* embed_b  = (const float*)d_in[2];
  const float* pre_W1   = (const float*)d_in[3];
  const float* pre_b1   = (const float*)d_in[4];
  const float* post_W1  = (const float*)d_in[5];
  const float* post_b1  = (const float*)d_in[6];
  const float* lin_W1   = (const float*)d_in[7];
  const float* lin_b1   = (const float*)d_in[8];
  const float* bn_gamma = (const float*)d_in[9];
  const float* bn_beta  = (const float*)d_in[10];
  const float* pre_W2   = (const float*)d_in[11];
  const float* pre_b2   = (const float*)d_in[12];
  const float* post_W2  = (const float*)d_in[13];
  const float* post_b2  = (const float*)d_in[14];
  const float* lin_W2   = (const float*)d_in[15];
  const float* lin_b2   = (const float*)d_in[16];
  const int*   src      = (const int*)d_in[17];
  const int*   dst      = (const int*)d_in[18];
  float* out = (float*)d_out;

  // Workspace layout (4-byte units); total ~81 MB -> L2 resident (192 MB)
  float* ws      = (float*)d_ws;
  int*   degi    = (int*)ws;                         // N
  float* misc    = ws + N_NODES;                     // [0]=logsum,[64..]=bnsum,[128..]=bnsq
  int*   row_ptr = (int*)(misc + 256);               // N+1
  int*   cursor  = row_ptr + N_NODES + 1;            // N
  int*   col     = cursor + N_NODES;                 // E
  float* h       = (float*)(col + E_EDGES);          // N*64
  float* hout    = h + (size_t)N_NODES * H_DIM;      // N*64
  float* Ad      = hout + (size_t)N_NODES * H_DIM;   // N*128
  float* Bs      = Ad + (size_t)N_NODES * 128;       // N*128

  const int tiles = N_NODES / 16;  // 3125
  const int eb = (E_EDGES + 255) / 256;
  const int bn_blocks = (N_NODES * H_DIM + 255) / 256;

  // Graph preprocessing: degree -> row_ptr (scan) -> CSR fill
  k_zero<<<256, 256, 0, stream>>>(ws, (size_t)N_NODES + 256);  // degi + misc
  k_degcount<<<eb, 256, 0, stream>>>(dst, degi);
  k_scan<<<1, 1024, 0, stream>>>(degi, row_ptr, cursor);
  k_fill<<<eb, 256, 0, stream>>>(src, dst, cursor, col);
  k_avglog<<<256, 256, 0, stream>>>(row_ptr, misc);

  k_embed<<<tiles, 128, 0, stream>>>(x, embed_W, embed_b, h);

  // ---- layer 1 ----
  k_pre<<<dim3(tiles, 4), 128, 0, stream>>>(h, pre_W1, pre_b1, Ad, Bs);
  k_post<<<tiles, 128, 0, stream>>>(h, row_ptr, col, misc, Ad, Bs,
                                    post_W1, post_b1, lin_W1, lin_b1, hout);
  // ---- batch norm + relu ----
  k_bnstat<<<512, 256, 0, stream>>>(hout, misc + 64, misc + 128);
  k_bnapply<<<bn_blocks, 256, 0, stream>>>(hout, misc + 64, misc + 128, bn_gamma,
                                           bn_beta, h);
  // ---- layer 2 ----
  k_pre<<<dim3(tiles, 4), 128, 0, stream>>>(h, pre_W2, pre_b2, Ad, Bs);
  k_post<<<tiles, 128, 0, stream>>>(h, row_ptr, col, misc, Ad, Bs,
                                    post_W2, post_b2, lin_W2, lin_b2, out);
}